// LogisticRegressionTorchModule_60876866454253
// MI455X (gfx1250) — compile-verified
//
#include <hip/hip_runtime.h>
#include <math.h>

typedef __attribute__((ext_vector_type(2))) float v2f;
typedef __attribute__((ext_vector_type(4))) float v4f;
typedef __attribute__((ext_vector_type(8))) float v8f;

#define N_ROWS 100000
#define DIM    512
#define NSAMP  128

// Kernel 0: ew[k] = exp(w_log_var[k]) into workspace (one tiny block).
__global__ void exp_precompute_kernel(const float* __restrict__ wlv,
                                      float* __restrict__ ew) {
    int i = threadIdx.x;
    if (i < DIM) ew[i] = __expf(wlv[i]);
}

// Main fused kernel: 512 threads = 16 waves; one 16-row tile per block.
__global__ __launch_bounds__(512) void bayes_linear_wmma_kernel(
    const float* __restrict__ x,
    const float* __restrict__ w_mu,
    const float* __restrict__ ew,
    const float* __restrict__ z,
    float* __restrict__ out)
{
    __shared__ float s_std[16];
    __shared__ float s_mean[16];

    const int tid  = threadIdx.x;
    const int wave = tid >> 5;
    const int lane = tid & 31;
    const int row0 = blockIdx.x << 4;          // 16-row tile base
    const int row  = row0 + wave;              // this wave's row

    // ---- Phase 1: dual dot products for one row, coalesced 16B loads ----
    // x is a 204.8 MB single-use stream -> non-temporal loads (keep L2 for the
    // broadcast operands w_mu/ew/z, which every block re-reads).
    const v4f* __restrict__ xr  = (const v4f*)(x + (size_t)row * DIM);
    const v4f* __restrict__ wm4 = (const v4f*)w_mu;
    const v4f* __restrict__ ew4 = (const v4f*)ew;

    float accm = 0.0f;   // sum x * w_mu
    float accv = 0.0f;   // sum x*x * exp(w_log_var)
    #pragma unroll
    for (int i = 0; i < DIM / 128; ++i) {      // 4 iterations, 512B/wave/load
        const int idx = lane + (i << 5);
        v4f xv = __builtin_nontemporal_load(&xr[idx]);
        v4f wv = wm4[idx];
        v4f ev = ew4[idx];
        #pragma unroll
        for (int c = 0; c < 4; ++c) {
            accm = fmaf(xv[c], wv[c], accm);
            accv = fmaf(xv[c] * xv[c], ev[c], accv);
        }
    }
    #pragma unroll
    for (int off = 16; off >= 1; off >>= 1) {  // wave32 tree reduction
        accm += __shfl_xor(accm, off, 32);
        accv += __shfl_xor(accv, off, 32);
    }
    if (lane == 0) {
        s_mean[wave] = accm;
        s_std[wave]  = sqrtf(accv);
    }
    __syncthreads();

    // ---- Phase 2: out_tile(16x16) = [std|mean|0|0](16x4) @ [z;1;0;0](4x16) ----
    // via V_WMMA_F32_16X16X4_F32 (full fp32 matrix op). Waves 0..7 cover NS=128.
    if (wave < 8) {
        const int  col0 = wave << 4;
        const bool lo   = lane < 16;
        const int  l15  = lane & 15;

        // A 16x4 layout: VGPR0 = K=0 (lanes 0-15) / K=2 (lanes 16-31),
        //                VGPR1 = K=1 / K=3. M = lane%16.
        v2f a;
        a[0] = lo ? s_std[l15]  : 0.0f;   // A[m][0] = std[m]
        a[1] = lo ? s_mean[l15] : 0.0f;   // A[m][1] = mean[m]
        // B 4x16 layout mirrors A: N = lane%16, K in {0,1} for lanes 0-15.
        v2f b;
        b[0] = lo ? z[col0 + l15] : 0.0f; // B[0][n] = z[n]
        b[1] = lo ? 1.0f          : 0.0f; // B[1][n] = 1
        v8f c = {};

        v8f d = __builtin_amdgcn_wmma_f32_16x16x4_f32(
            /*neg_a=*/false, a, /*neg_b=*/false, b,
            /*c_mod=*/(short)0, c, /*reuse_a=*/false, /*reuse_b=*/false);

        // C/D layout: VGPR r holds M = r + 8*(lane>=16), N = lane&15.
        // Output is a 51.2 MB single-use stream -> non-temporal stores.
        float* __restrict__ obase = out + (size_t)row0 * NSAMP + col0;
        const int m_hi = (lane >> 4) << 3;
        #pragma unroll
        for (int r = 0; r < 8; ++r) {
            __builtin_nontemporal_store(d[r],
                &obase[(size_t)(r + m_hi) * NSAMP + l15]);
        }
    }
}

extern "C" void kernel_launch(void* const* d_in, const int* in_sizes, int n_in,
                              void* d_out, int out_size, void* d_ws, size_t ws_size,
                              hipStream_t stream) {
    const float* x    = (const float*)d_in[0];   // [N, D]
    const float* w_mu = (const float*)d_in[1];   // [D]
    const float* wlv  = (const float*)d_in[2];   // [D]
    const float* z    = (const float*)d_in[3];   // [NS]
    float*       out  = (float*)d_out;           // [N, NS]
    float*       ew   = (float*)d_ws;            // [D] scratch: exp(w_log_var)

    exp_precompute_kernel<<<1, DIM, 0, stream>>>(wlv, ew);
    bayes_linear_wmma_kernel<<<N_ROWS / 16, 512, 0, stream>>>(x, w_mu, ew, z, out);
}